// QueryCentricNet_28132035789166
// MI455X (gfx1250) — compile-verified
//
#include <hip/hip_runtime.h>
#include <hip/hip_bf16.h>

#define WINR 7
#define WW 15
#define MAXM 50
#define DQ 32
#define DD 32
#define EE 300
#define QLEN 32
#define DLEN 8192
#define PADLEN (DLEN + 2 * WINR)            // 8206
#define CPW (DQ + DD + 1)                   // 65
#define BLK_ELEMS (QLEN * WW * CPW)         // 31200 floats per (q,m) block
#define BLK_VEC4 (BLK_ELEMS / 4)            // 7800 float4 per block
#define FUSED_ELEMS ((size_t)QLEN * MAXM * BLK_ELEMS)  // 49,920,000

typedef __attribute__((ext_vector_type(2))) float v2f;
typedef __attribute__((ext_vector_type(8))) float v8f;

// ---------------- kernel 1: rank-sort the 32 distinct query ids ----------------
__global__ void k_sort(const int* __restrict__ q, int* __restrict__ qs) {
    __shared__ int s[QLEN];
    int t = threadIdx.x;
    s[t] = q[t];
    __syncthreads();
    int v = s[t], rank = 0;
    for (int j = 0; j < QLEN; ++j) rank += (s[j] < v) ? 1 : 0;
    qs[rank] = v;
}

// ---------------- kernel 2: first-50 match positions per query (ordered) -------
__global__ void k_match(const int* __restrict__ d, const int* __restrict__ qs,
                        int* __restrict__ pos, int* __restrict__ kcnt) {
    int qi = blockIdx.x;
    int lane = threadIdx.x;               // one wave32 per query
    int qval = qs[qi];
    int base = 0;
    for (int c = 0; c < DLEN / 32; ++c) {
        if (base >= MAXM) break;
        int idx = c * 32 + lane;
        bool m = (d[idx] == qval);
        unsigned mask = (unsigned)__ballot(m);    // wave32: low 32 bits
        if (m) {
            int r = base + __popc(mask & ((1u << lane) - 1u));
            if (r < MAXM) pos[qi * MAXM + r] = idx + WINR;  // padded coords
        }
        base += __popc(mask);
    }
    if (lane == 0) kcnt[qi] = base < MAXM ? base : MAXM;
}

// ---------------- kernel 3: q_embed gather + q_comp = q_embed @ wq^T + bq ------
__global__ void __launch_bounds__(256) k_qcomp(
    const float* __restrict__ emb, const int* __restrict__ qs,
    const float* __restrict__ wq, const float* __restrict__ bq,
    float* __restrict__ qembed, float* __restrict__ qcomp) {
    __shared__ float s_qe[QLEN * EE];
    int tid = threadIdx.x;
    for (int i = tid; i < QLEN * EE; i += 256) {
        int row = i / EE, col = i - row * EE;
        float v = emb[(size_t)qs[row] * EE + col];
        s_qe[i] = v;
        qembed[i] = v;
    }
    __syncthreads();
    for (int o = tid; o < QLEN * DQ; o += 256) {
        int x = o / DQ, c = o - x * DQ;
        float acc = bq[c];
        const float* a = &s_qe[x * EE];
        const float* b = &wq[c * EE];
        for (int k = 0; k < EE; ++k) acc = fmaf(a[k], b[k], acc);
        qcomp[o] = acc;
    }
}

// ---------------- kernel 4: per-(q,m) fused block --------------------------------
// waves 0,1: interact = q_embed[32,300] @ win_embed^T  -> [32,15]  (2 M-tiles)
// waves 2,3: win_comp = win_embed[15,300] @ wd^T + bd  -> [15,32]  (2 N-tiles)
__global__ void __launch_bounds__(128) k_fuse(
    const int* __restrict__ d, const float* __restrict__ emb,
    const float* __restrict__ wd, const float* __restrict__ bd,
    const int* __restrict__ pos, const int* __restrict__ kcnt,
    const float* __restrict__ qembed, const float* __restrict__ qcomp,
    float* __restrict__ out) {
    __shared__ float s_win[16][EE];       // row 15 zero-padded
    __shared__ float s_itc[QLEN][16];     // interact [x][w] (col 15 unused)
    __shared__ float s_wc[16][DD];        // win_comp [w][dd] (row 15 unused)
    __shared__ float s_qc[QLEN * DQ];
    __shared__ int   s_tok[WW];

    int b = blockIdx.x;
    int qi = b / MAXM;
    int m  = b - qi * MAXM;
    int tid = threadIdx.x;
    size_t base = (size_t)b * BLK_ELEMS;
    float4* __restrict__ out4 = (float4*)(out + base);  // block base is 16B aligned

    bool valid = (m < kcnt[qi]);
    if (tid == 0) out[FUSED_ELEMS + (size_t)b] = valid ? 1.0f : 0.0f;

    if (!valid) {  // masked-out slot: fused block is all zeros (b128 stores)
        float4 z; z.x = z.y = z.z = z.w = 0.0f;
        for (int j = tid; j < BLK_VEC4; j += 128) out4[j] = z;
        return;
    }

    // window tokens around the match position (padded coords, clip + pad -> -1)
    if (tid < WW) {
        int p = pos[b] + tid - WINR;
        p = p < 0 ? 0 : (p > PADLEN - 1 ? PADLEN - 1 : p);
        s_tok[tid] = (p < WINR || p >= WINR + DLEN) ? -1 : d[p - WINR];
    }
    for (int i = tid; i < QLEN * DQ; i += 128) s_qc[i] = qcomp[i];
    __syncthreads();

    // gather win_embed into LDS; pad token / pad row -> zeros
    for (int i = tid; i < 16 * EE; i += 128) {
        int row = i / EE, col = i - row * EE;
        float v = 0.0f;
        if (row < WW) {
            int tok = s_tok[row];
            if (tok >= 0) v = emb[(size_t)tok * EE + col];
        }
        s_win[row][col] = v;
    }
    __syncthreads();

    int wave  = tid >> 5;
    int lane  = tid & 31;
    int nloc  = lane & 15;
    int khalf = lane >> 4;       // lanes 16..31 hold K+2,K+3
    v8f acc = {};

    if (wave < 2) {
        // C[x][w]: A row = q_embed[x], B col w = win_embed[w][k]
        const float* arow = &qembed[(wave * 16 + nloc) * EE];
        const float* brow = &s_win[nloc][0];
        for (int kk = 0; kk < EE; kk += 4) {
            int k = kk + khalf * 2;
            v2f a = { arow[k], arow[k + 1] };
            v2f bb = { brow[k], brow[k + 1] };
            acc = __builtin_amdgcn_wmma_f32_16x16x4_f32(
                false, a, false, bb, (short)0, acc, false, false);
        }
        for (int r = 0; r < 8; ++r) {
            int M = r + (khalf ? 8 : 0);     // x within tile
            int w = nloc;
            if (w < WW) s_itc[wave * 16 + M][w] = acc[r];
        }
    } else {
        // C[w][dd]: A row = win_embed[w], B col dd = wd[dd][k]
        int t = wave - 2;
        const float* arow = &s_win[nloc][0];
        const float* brow = &wd[(t * 16 + nloc) * EE];
        for (int kk = 0; kk < EE; kk += 4) {
            int k = kk + khalf * 2;
            v2f a = { arow[k], arow[k + 1] };
            v2f bb = { brow[k], brow[k + 1] };
            acc = __builtin_amdgcn_wmma_f32_16x16x4_f32(
                false, a, false, bb, (short)0, acc, false, false);
        }
        float bias = bd[t * 16 + nloc];
        for (int r = 0; r < 8; ++r) {
            int M = r + (khalf ? 8 : 0);     // w
            if (M < WW) s_wc[M][t * 16 + nloc] = acc[r] + bias;
        }
    }
    __syncthreads();

    // stream the [32,15,65] block with b128 stores:
    // c in [0:32)=q_comp[x][c], [32:64)=win_comp[w][c-32], [64]=interact[x][w]
    auto fetch = [&](int p) -> float {
        int x  = p / (WW * CPW);
        int r2 = p - x * (WW * CPW);
        int w  = r2 / CPW;
        int c  = r2 - w * CPW;
        if (c < DQ)           return s_qc[x * DQ + c];
        else if (c < DQ + DD) return s_wc[w][c - DQ];
        else                  return s_itc[x][w];
    };
    for (int j = tid; j < BLK_VEC4; j += 128) {
        int p = 4 * j;
        float4 v;
        v.x = fetch(p);
        v.y = fetch(p + 1);
        v.z = fetch(p + 2);
        v.w = fetch(p + 3);
        out4[j] = v;
    }
}

extern "C" void kernel_launch(void* const* d_in, const int* in_sizes, int n_in,
                              void* d_out, int out_size, void* d_ws, size_t ws_size,
                              hipStream_t stream) {
    const int*   q   = (const int*)d_in[0];
    const int*   d   = (const int*)d_in[1];
    const float* emb = (const float*)d_in[2];
    const float* wq  = (const float*)d_in[3];
    const float* bq  = (const float*)d_in[4];
    const float* wd  = (const float*)d_in[5];
    const float* bd  = (const float*)d_in[6];
    float* out = (float*)d_out;

    // workspace layout
    char* ws = (char*)d_ws;
    int*   ws_qs     = (int*)ws;                       ws += 64 * sizeof(int);
    int*   ws_pos    = (int*)ws;                       ws += QLEN * MAXM * sizeof(int);
    int*   ws_k      = (int*)ws;                       ws += 64 * sizeof(int);
    float* ws_qembed = (float*)ws;                     ws += QLEN * EE * sizeof(float);
    float* ws_qcomp  = (float*)ws;

    k_sort <<<1, QLEN, 0, stream>>>(q, ws_qs);
    k_match<<<QLEN, 32, 0, stream>>>(d, ws_qs, ws_pos, ws_k);
    k_qcomp<<<1, 256, 0, stream>>>(emb, ws_qs, wq, bq, ws_qembed, ws_qcomp);
    k_fuse <<<QLEN * MAXM, 128, 0, stream>>>(d, emb, wd, bd, ws_pos, ws_k,
                                             ws_qembed, ws_qcomp, out);
}